// GCN_70428873720075
// MI455X (gfx1250) — compile-verified
//
#include <hip/hip_runtime.h>
#include <hip/hip_bf16.h>
#include <math.h>

typedef __attribute__((ext_vector_type(2))) float v2f;
typedef __attribute__((ext_vector_type(8))) float v8f;

#define N_FEAT 128   // K of GEMM1 (fixed by reference)
#define HID    16    // hidden width == WMMA tile N
#define NCLS   10    // classes (padded to 16 in GEMM2)

// ---------------- utility ----------------
__global__ void zero_f(float* __restrict__ p, int n) {
  int i = blockIdx.x * blockDim.x + threadIdx.x;
  if (i < n) p[i] = 0.0f;
}

// ---------------- degree / normalization ----------------
__global__ void deg_scatter(const long long* __restrict__ ei, float* __restrict__ deg,
                            int nEdges) {
  int e = blockIdx.x * blockDim.x + threadIdx.x;
  if (e >= nEdges) return;
  int d = (int)ei[(size_t)nEdges + e];   // dst row of edge_index
  atomicAdd(&deg[d], 1.0f);
}

__global__ void dinv_finalize(float* __restrict__ deg, int n) {
  int i = blockIdx.x * blockDim.x + threadIdx.x;
  if (i < n) deg[i] = rsqrtf(deg[i] + 1.0f);   // +1 self-loop
}

// ---------------- GEMM1: h1[N,16] = x[N,128] @ W1[128,16] (fp32 WMMA) ----------------
__global__ void gemm1_wmma(const float* __restrict__ x, const float* __restrict__ W1,
                           float* __restrict__ h1, int nNodes) {
  const int lane = threadIdx.x & 31;
  const int wave = blockIdx.x * (blockDim.x >> 5) + (threadIdx.x >> 5);
  const int row0 = wave * 16;
  if (row0 >= nNodes) return;              // wave-uniform: EXEC stays all-1 for WMMA
  const bool full = (row0 + 16 <= nNodes); // wave-uniform (N=100000 -> always true)
  const int hi = lane >> 4;                // which half-wave (K split)
  const int lo = lane & 15;                // M for A, N for B/C
  int m = row0 + lo;
  if (!full && m > nNodes - 1) m = nNodes - 1;   // clamp only in ragged tail
  const float* xr = x + (size_t)m * N_FEAT;
  v8f c = {};
#pragma unroll 8
  for (int k = 0; k < N_FEAT; k += 4) {
    const int ka = k + 2 * hi;
    v2f a; a.x = xr[ka];                    a.y = xr[ka + 1];
    v2f b; b.x = W1[(size_t)ka * HID + lo]; b.y = W1[(size_t)(ka + 1) * HID + lo];
    c = __builtin_amdgcn_wmma_f32_16x16x4_f32(false, a, false, b, (short)0, c,
                                              false, false);
  }
  // C/D layout: VGPR v -> row M = v + 8*hi, column N = lo
  if (full) {
    float* hp = h1 + (size_t)(row0 + 8 * hi) * HID + lo;
#pragma unroll
    for (int v = 0; v < 8; ++v) hp[(size_t)v * HID] = c[v];  // 8 stores, imm offsets
  } else {
#pragma unroll
    for (int v = 0; v < 8; ++v) {
      const int r = row0 + v + 8 * hi;
      if (r < nNodes) h1[(size_t)r * HID + lo] = c[v];
    }
  }
}

// ---------------- GEMM2: h2[N,16] = h[N,16] @ W2pad[16,16] (fp32 WMMA) ----------------
__global__ void gemm2_wmma(const float* __restrict__ h, const float* __restrict__ W2,
                           float* __restrict__ h2, int nNodes) {
  const int lane = threadIdx.x & 31;
  const int wave = blockIdx.x * (blockDim.x >> 5) + (threadIdx.x >> 5);
  const int row0 = wave * 16;
  if (row0 >= nNodes) return;
  const bool full = (row0 + 16 <= nNodes);
  const int hi = lane >> 4;
  const int lo = lane & 15;
  int m = row0 + lo;
  if (!full && m > nNodes - 1) m = nNodes - 1;
  const float* hr = h + (size_t)m * HID;
  v8f c = {};
#pragma unroll
  for (int k = 0; k < HID; k += 4) {
    const int ka = k + 2 * hi;
    v2f a; a.x = hr[ka]; a.y = hr[ka + 1];
    v2f b;                                   // W2 is [16,10]; pad cols 10..15 with 0
    b.x = (lo < NCLS) ? W2[(size_t)ka * NCLS + lo]       : 0.0f;
    b.y = (lo < NCLS) ? W2[(size_t)(ka + 1) * NCLS + lo] : 0.0f;
    c = __builtin_amdgcn_wmma_f32_16x16x4_f32(false, a, false, b, (short)0, c,
                                              false, false);
  }
  if (full) {
    float* hp = h2 + (size_t)(row0 + 8 * hi) * HID + lo;
#pragma unroll
    for (int v = 0; v < 8; ++v) hp[(size_t)v * HID] = c[v];  // cols >= 10 exact zeros
  } else {
#pragma unroll
    for (int v = 0; v < 8; ++v) {
      const int r = row0 + v + 8 * hi;
      if (r < nNodes) h2[(size_t)r * HID + lo] = c[v];
    }
  }
}

// ---------------- edge aggregation: agg[dst] += h[src] * dinv[src]*dinv[dst] ----------------
__global__ void agg_scatter(const float* __restrict__ hsrc, const long long* __restrict__ ei,
                            const float* __restrict__ dinv, float* __restrict__ agg,
                            int nEdges) {
  int e = blockIdx.x * blockDim.x + threadIdx.x;
  if (e >= nEdges) return;
  int s = (int)ei[e];
  int d = (int)ei[(size_t)nEdges + e];
  float coef = dinv[s] * dinv[d];
  const float4* hp = (const float4*)(hsrc + (size_t)s * HID);
  float* ap = agg + (size_t)d * HID;
#pragma unroll
  for (int q = 0; q < 4; ++q) {
    float4 v = hp[q];
    atomicAdd(ap + 4 * q + 0, v.x * coef);
    atomicAdd(ap + 4 * q + 1, v.y * coef);
    atomicAdd(ap + 4 * q + 2, v.z * coef);
    atomicAdd(ap + 4 * q + 3, v.w * coef);
  }
}

// ---------------- self-loop + bias (+relu for layer 1) ----------------
__global__ void selfloop_relu(float* __restrict__ agg, const float* __restrict__ h,
                              const float* __restrict__ bias, const float* __restrict__ dinv,
                              int nNodes) {
  int idx = blockIdx.x * blockDim.x + threadIdx.x;
  if (idx >= nNodes * HID) return;
  int i = idx >> 4, c = idx & 15;
  float di = dinv[i];
  float v = agg[idx] + h[idx] * di * di + bias[c];
  agg[idx] = fmaxf(v, 0.0f);
}

__global__ void selfloop2(float* __restrict__ agg, const float* __restrict__ h,
                          const float* __restrict__ b2, const float* __restrict__ dinv,
                          int nNodes) {
  int idx = blockIdx.x * blockDim.x + threadIdx.x;
  if (idx >= nNodes * HID) return;
  int i = idx >> 4, c = idx & 15;
  float di = dinv[i];
  float bc = (c < NCLS) ? b2[c] : 0.0f;
  agg[idx] = agg[idx] + h[idx] * di * di + bc;
}

// ---------------- global mean pool + log_softmax ----------------
__global__ void pool_scatter(const float* __restrict__ h, const long long* __restrict__ batch,
                             float* __restrict__ gsum, float* __restrict__ gcnt, int nNodes) {
  int i = blockIdx.x * blockDim.x + threadIdx.x;
  if (i >= nNodes) return;
  int g = (int)batch[i];
  atomicAdd(&gcnt[g], 1.0f);
  const float* hr = h + (size_t)i * HID;
#pragma unroll
  for (int c = 0; c < NCLS; ++c) atomicAdd(&gsum[(size_t)g * HID + c], hr[c]);
}

__global__ void pool_finalize(const float* __restrict__ gsum, const float* __restrict__ gcnt,
                              float* __restrict__ out, int nGraphs) {
  int g = blockIdx.x * blockDim.x + threadIdx.x;
  if (g >= nGraphs) return;
  float cnt = fmaxf(gcnt[g], 1.0f);
  float v[NCLS];
  float m = -1e30f;
#pragma unroll
  for (int c = 0; c < NCLS; ++c) { v[c] = gsum[(size_t)g * HID + c] / cnt; m = fmaxf(m, v[c]); }
  float s = 0.0f;
#pragma unroll
  for (int c = 0; c < NCLS; ++c) s += expf(v[c] - m);
  float lse = logf(s);
#pragma unroll
  for (int c = 0; c < NCLS; ++c) out[(size_t)g * NCLS + c] = v[c] - m - lse;
}

// ---------------- driver ----------------
extern "C" void kernel_launch(void* const* d_in, const int* in_sizes, int n_in,
                              void* d_out, int out_size, void* d_ws, size_t ws_size,
                              hipStream_t stream) {
  const float*     x     = (const float*)d_in[0];
  const long long* ei    = (const long long*)d_in[1];   // int64 [2, E]
  const long long* batch = (const long long*)d_in[2];   // int64 [N]
  const float*     W1    = (const float*)d_in[3];
  const float*     b1    = (const float*)d_in[4];
  const float*     W2    = (const float*)d_in[5];
  const float*     b2    = (const float*)d_in[6];
  float*           out   = (float*)d_out;

  const int nNodes  = in_sizes[0] / N_FEAT;
  const int nEdges  = in_sizes[1] / 2;
  const int nGraphs = out_size / NCLS;

  // scratch layout (floats): dinv[N] | h1/h2[16N] | hbuf[16N] | gsum[G*16] | gcnt[G]
  float* dinv = (float*)d_ws;
  float* h1   = dinv + nNodes;
  float* hbuf = h1 + (size_t)nNodes * HID;
  float* gsum = hbuf + (size_t)nNodes * HID;
  float* gcnt = gsum + (size_t)nGraphs * HID;

  const int T = 256;
  auto cdiv = [](int a, int b) { return (a + b - 1) / b; };
  const int tiles = cdiv(nNodes, 16);
  const int gemmBlocks = cdiv(tiles, 4);   // 4 waves (128 threads) per block

  // degrees -> dinv
  zero_f<<<cdiv(nNodes, T), T, 0, stream>>>(dinv, nNodes);
  deg_scatter<<<cdiv(nEdges, T), T, 0, stream>>>(ei, dinv, nEdges);
  dinv_finalize<<<cdiv(nNodes, T), T, 0, stream>>>(dinv, nNodes);

  // layer 1: h1 = x@W1 ; hbuf = relu(scatter(h1) + selfloop + b1)
  gemm1_wmma<<<gemmBlocks, 128, 0, stream>>>(x, W1, h1, nNodes);
  zero_f<<<cdiv(nNodes * HID, T), T, 0, stream>>>(hbuf, nNodes * HID);
  agg_scatter<<<cdiv(nEdges, T), T, 0, stream>>>(h1, ei, dinv, hbuf, nEdges);
  selfloop_relu<<<cdiv(nNodes * HID, T), T, 0, stream>>>(hbuf, h1, b1, dinv, nNodes);

  // layer 2: h2 (reusing h1 buffer) = hbuf@W2pad ; hbuf reused as agg2/result
  gemm2_wmma<<<gemmBlocks, 128, 0, stream>>>(hbuf, W2, h1, nNodes);
  zero_f<<<cdiv(nNodes * HID + nGraphs * HID + nGraphs, T), T, 0, stream>>>(
      hbuf, nNodes * HID + nGraphs * HID + nGraphs);   // also zeroes gsum/gcnt
  agg_scatter<<<cdiv(nEdges, T), T, 0, stream>>>(h1, ei, dinv, hbuf, nEdges);
  selfloop2<<<cdiv(nNodes * HID, T), T, 0, stream>>>(hbuf, h1, b2, dinv, nNodes);

  // pooling + log_softmax
  pool_scatter<<<cdiv(nNodes, T), T, 0, stream>>>(hbuf, batch, gsum, gcnt, nNodes);
  pool_finalize<<<1, 64, 0, stream>>>(gsum, gcnt, out, nGraphs);
}